// HAN_65712999628924
// MI455X (gfx1250) — compile-verified
//
#include <hip/hip_runtime.h>
#include <climits>

typedef float v2f __attribute__((ext_vector_type(2)));
typedef float v8f __attribute__((ext_vector_type(8)));

#define N_GENE 100000
#define N_PATH 20000
#define IN_CH 128
#define HID 128
#define HEADS 8
#define DHEAD 16
#define OUT_CH 2
#define E_G2G 600000
#define E_P2G 400000

// ---------- ordered-int encoding for float atomic max ----------
__device__ __forceinline__ int f2ord(float f) {
    int b = __float_as_int(f);
    return b >= 0 ? b : (b ^ 0x7FFFFFFF);
}
__device__ __forceinline__ float ord2f(int k) {
    return __int_as_float(k >= 0 ? k : (k ^ 0x7FFFFFFF));
}

// =====================================================================
// Projection GEMM: H[r, c] = sum_k X[r,k] * W[c,k] + bias[c]
// One wave per 16x16 output tile, V_WMMA_F32_16X16X4_F32, K=128 in 32 steps.
// A frag (16x4): lane l holds A[l&15][koff+j], koff = (l>=16)?2:0, j=0,1
// B frag (4x16): lane l holds B[koff+j][l&15] = W[l&15 + col0][k0+koff+j]
// C/D (16x16):  vgpr v -> row v + (l>=16 ? 8 : 0), col l&15
// =====================================================================
__global__ void proj_wmma_kernel(const float* __restrict__ X,
                                 const float* __restrict__ W,
                                 const float* __restrict__ bias,
                                 float* __restrict__ H) {
    const int wave = threadIdx.x >> 5;         // 8 waves: 8 column tiles of 16
    const int lane = threadIdx.x & 31;
    const int row0 = blockIdx.x * 16;
    const int col0 = wave * 16;
    const int mn   = lane & 15;
    const int koff = (lane >> 4) << 1;         // 0 or 2
    const float* xrow = X + (size_t)(row0 + mn) * IN_CH + koff;
    const float* wrow = W + (size_t)(col0 + mn) * IN_CH + koff;
    v8f acc = {};
#pragma unroll
    for (int k0 = 0; k0 < IN_CH; k0 += 4) {
        v2f a = *(const v2f*)(xrow + k0);
        v2f b = *(const v2f*)(wrow + k0);
        acc = __builtin_amdgcn_wmma_f32_16x16x4_f32(false, a, false, b,
                                                    (short)0, acc, false, false);
    }
    const float bv    = bias[col0 + mn];
    const int   mbase = row0 + ((lane >> 4) << 3);
#pragma unroll
    for (int v = 0; v < 8; ++v)
        H[(size_t)(mbase + v) * HID + col0 + mn] = acc[v] + bv;
}

// =====================================================================
// Semantic GEMM: per metapath, wsum[c] += sum_rows tanh( relu(Out)@Wk^T + bk )
// Same WMMA tiling; relu folded into the A loads, tanh+row-reduce on the tile.
// =====================================================================
__global__ void sem_wmma_kernel(const float* __restrict__ OutM,
                                const float* __restrict__ Wk,
                                const float* __restrict__ bk,
                                float* __restrict__ wsum) {
    const int wave = threadIdx.x >> 5;
    const int lane = threadIdx.x & 31;
    const int row0 = blockIdx.x * 16;
    const int col0 = wave * 16;
    const int mn   = lane & 15;
    const int koff = (lane >> 4) << 1;
    const float* arow = OutM + (size_t)(row0 + mn) * HID + koff;
    const float* wrow = Wk   + (size_t)(col0 + mn) * HID + koff;
    v8f acc = {};
#pragma unroll
    for (int k0 = 0; k0 < HID; k0 += 4) {
        v2f a = *(const v2f*)(arow + k0);
        a[0] = fmaxf(a[0], 0.0f);              // relu(stack input)
        a[1] = fmaxf(a[1], 0.0f);
        v2f b = *(const v2f*)(wrow + k0);
        acc = __builtin_amdgcn_wmma_f32_16x16x4_f32(false, a, false, b,
                                                    (short)0, acc, false, false);
    }
    const float bv = bk[col0 + mn];
    float partial = 0.0f;
#pragma unroll
    for (int v = 0; v < 8; ++v) partial += tanhf(acc[v] + bv);
    atomicAdd(&wsum[col0 + mn], partial);      // sum over this tile's 8 rows
}

// =====================================================================
// Per-(node,head) attention coefficient: out[n,h] = <H[n,h,:], att[h,:]>
// =====================================================================
__global__ void attn_coef_kernel(const float* __restrict__ H,
                                 const float* __restrict__ att,
                                 float* __restrict__ out, int nnodes) {
    int idx = blockIdx.x * blockDim.x + threadIdx.x;
    if (idx >= nnodes * HEADS) return;
    int node = idx >> 3, h = idx & 7;
    const float* hp = H + (size_t)node * HID + h * DHEAD;
    const float* ap = att + h * DHEAD;
    float s = 0.0f;
#pragma unroll
    for (int d = 0; d < DHEAD; ++d) s += hp[d] * ap[d];
    out[idx] = s;
}

__global__ void fill_i32_kernel(int* __restrict__ p, int v, int n) {
    int i = blockIdx.x * blockDim.x + threadIdx.x;
    if (i < n) p[i] = v;
}

// Pass 1: segment max of leaky_relu(a_s[src]+a_d[dst]) per (dst, head)
__global__ void edge_max_kernel(const int* __restrict__ src, const int* __restrict__ dst,
                                const float* __restrict__ as_, const float* __restrict__ ad_,
                                int* __restrict__ maxk, int nedge) {
    int idx = blockIdx.x * blockDim.x + threadIdx.x;
    if (idx >= nedge * HEADS) return;
    int e = idx >> 3, h = idx & 7;
    int s = src[e], d = dst[e];
    float v = as_[s * HEADS + h] + ad_[d * HEADS + h];
    v = v > 0.0f ? v : 0.2f * v;
    atomicMax(&maxk[d * HEADS + h], f2ord(v));
}

// Pass 2: denom[dst,h] += exp(e - m[dst,h])
__global__ void edge_den_kernel(const int* __restrict__ src, const int* __restrict__ dst,
                                const float* __restrict__ as_, const float* __restrict__ ad_,
                                const int* __restrict__ maxk, float* __restrict__ den,
                                int nedge) {
    int idx = blockIdx.x * blockDim.x + threadIdx.x;
    if (idx >= nedge * HEADS) return;
    int e = idx >> 3, h = idx & 7;
    int s = src[e], d = dst[e];
    float v = as_[s * HEADS + h] + ad_[d * HEADS + h];
    v = v > 0.0f ? v : 0.2f * v;
    int k  = maxk[d * HEADS + h];
    float m = (k == INT_MIN) ? 0.0f : ord2f(k);
    atomicAdd(&den[d * HEADS + h], expf(v - m));
}

// Pass 3: out[dst, :] += alpha[h] * Hsrc[src, :]  (128 threads = 128 channels)
__global__ void edge_scatter_kernel(const int* __restrict__ src, const int* __restrict__ dst,
                                    const float* __restrict__ as_, const float* __restrict__ ad_,
                                    const int* __restrict__ maxk, const float* __restrict__ den,
                                    const float* __restrict__ Hsrc, float* __restrict__ out) {
    __shared__ float alpha[HEADS];
    const int e = blockIdx.x;
    const int t = threadIdx.x;                 // 0..127
    const int s = src[e], d = dst[e];
    if (t < HEADS) {
        float v = as_[s * HEADS + t] + ad_[d * HEADS + t];
        v = v > 0.0f ? v : 0.2f * v;
        int k   = maxk[d * HEADS + t];
        float m = (k == INT_MIN) ? 0.0f : ord2f(k);
        float p = expf(v - m);
        float dn = den[d * HEADS + t];
        alpha[t] = dn > 0.0f ? p / dn : 0.0f;
    }
    __syncthreads();
    float a = alpha[t >> 4];
    atomicAdd(&out[(size_t)d * HID + t], Hsrc[(size_t)s * HID + t] * a);
}

// beta = softmax_m( <q, mean_n tanh(...)> ), wsum holds the tanh column sums
__global__ void beta_kernel(const float* __restrict__ wsum,
                            const float* __restrict__ q,
                            float* __restrict__ beta) {
    float s0 = 0.0f, s1 = 0.0f;
    const float inv = 1.0f / (float)N_GENE;
    for (int i = 0; i < HID; ++i) {
        s0 += q[i] * (wsum[i] * inv);
        s1 += q[i] * (wsum[HID + i] * inv);
    }
    float mx = fmaxf(s0, s1);
    float e0 = expf(s0 - mx), e1 = expf(s1 - mx);
    float r  = 1.0f / (e0 + e1);
    beta[0] = e0 * r;
    beta[1] = e1 * r;
}

// fused = beta0*relu(out_g2g) + beta1*relu(out_p2g); out = fused @ W_out^T + b_out
__global__ void final_kernel(const float* __restrict__ outg, const float* __restrict__ outp,
                             const float* __restrict__ beta,
                             const float* __restrict__ Wout, const float* __restrict__ bout,
                             float* __restrict__ out) {
    int n = blockIdx.x * blockDim.x + threadIdx.x;
    if (n >= N_GENE) return;
    const float b0 = beta[0], b1 = beta[1];
    const float* g = outg + (size_t)n * HID;
    const float* p = outp + (size_t)n * HID;
    float acc0 = 0.0f, acc1 = 0.0f;
#pragma unroll 4
    for (int c = 0; c < HID; ++c) {
        float f = b0 * fmaxf(g[c], 0.0f) + b1 * fmaxf(p[c], 0.0f);
        acc0 += f * Wout[c];
        acc1 += f * Wout[HID + c];
    }
    out[n * OUT_CH + 0] = acc0 + bout[0];
    out[n * OUT_CH + 1] = acc1 + bout[1];
}

extern "C" void kernel_launch(void* const* d_in, const int* in_sizes, int n_in,
                              void* d_out, int out_size, void* d_ws, size_t ws_size,
                              hipStream_t stream) {
    const float* x_gene      = (const float*)d_in[0];
    const float* x_path      = (const float*)d_in[1];
    const int*   g2g_src     = (const int*)d_in[2];
    const int*   g2g_dst     = (const int*)d_in[2] + E_G2G;
    const int*   p2g_src     = (const int*)d_in[3];
    const int*   p2g_dst     = (const int*)d_in[4];
    const float* W_pg        = (const float*)d_in[5];
    const float* b_pg        = (const float*)d_in[6];
    const float* W_pp        = (const float*)d_in[7];
    const float* b_pp        = (const float*)d_in[8];
    const float* att_src_g2g = (const float*)d_in[9];
    const float* att_dst_g2g = (const float*)d_in[10];
    const float* att_src_p2g = (const float*)d_in[11];
    const float* att_dst_p2g = (const float*)d_in[12];
    const float* q           = (const float*)d_in[13];
    const float* Wk          = (const float*)d_in[14];
    const float* bk          = (const float*)d_in[15];
    const float* W_out       = (const float*)d_in[16];
    const float* b_out       = (const float*)d_in[17];
    float*       out         = (float*)d_out;

    // ---- workspace carve (~187 MB) ----
    float* ws = (float*)d_ws;
    size_t off = 0;
    auto carve = [&](size_t n) { float* p = ws + off; off += n; return p; };
    float* h_gene  = carve((size_t)N_GENE * HID);
    float* h_path  = carve((size_t)N_PATH * HID);
    float* out_g2g = carve((size_t)N_GENE * HID);
    float* out_p2g = carve((size_t)N_GENE * HID);
    float* as_g2g  = carve((size_t)N_GENE * HEADS);
    float* ad_g2g  = carve((size_t)N_GENE * HEADS);
    float* as_p2g  = carve((size_t)N_PATH * HEADS);
    float* ad_p2g  = carve((size_t)N_GENE * HEADS);
    int*   mk_g2g  = (int*)carve((size_t)N_GENE * HEADS);
    int*   mk_p2g  = (int*)carve((size_t)N_GENE * HEADS);
    float* den_g2g = carve((size_t)N_GENE * HEADS);
    float* den_p2g = carve((size_t)N_GENE * HEADS);
    float* wsum    = carve(2 * HID);
    float* beta    = carve(2);

    // ---- init (graph-capture safe) ----
    hipMemsetAsync(out_g2g, 0, (size_t)N_GENE * HID * sizeof(float), stream);
    hipMemsetAsync(out_p2g, 0, (size_t)N_GENE * HID * sizeof(float), stream);
    hipMemsetAsync(den_g2g, 0, (size_t)N_GENE * HEADS * sizeof(float), stream);
    hipMemsetAsync(den_p2g, 0, (size_t)N_GENE * HEADS * sizeof(float), stream);
    hipMemsetAsync(wsum,    0, 2 * HID * sizeof(float), stream);
    {
        int n = N_GENE * HEADS, blks = (n + 255) / 256;
        fill_i32_kernel<<<blks, 256, 0, stream>>>(mk_g2g, INT_MIN, n);
        fill_i32_kernel<<<blks, 256, 0, stream>>>(mk_p2g, INT_MIN, n);
    }

    // ---- projections (WMMA f32) ----
    proj_wmma_kernel<<<N_GENE / 16, 256, 0, stream>>>(x_gene, W_pg, b_pg, h_gene);
    proj_wmma_kernel<<<N_PATH / 16, 256, 0, stream>>>(x_path, W_pp, b_pp, h_path);

    // ---- per-node attention coefficients ----
    {
        int ng = N_GENE * HEADS, bg = (ng + 255) / 256;
        int np = N_PATH * HEADS, bp = (np + 255) / 256;
        attn_coef_kernel<<<bg, 256, 0, stream>>>(h_gene, att_src_g2g, as_g2g, N_GENE);
        attn_coef_kernel<<<bg, 256, 0, stream>>>(h_gene, att_dst_g2g, ad_g2g, N_GENE);
        attn_coef_kernel<<<bp, 256, 0, stream>>>(h_path, att_src_p2g, as_p2g, N_PATH);
        attn_coef_kernel<<<bg, 256, 0, stream>>>(h_gene, att_dst_p2g, ad_p2g, N_GENE);
    }

    // ---- edge softmax: max / denom / scatter ----
    {
        int ng = E_G2G * HEADS, bg = (ng + 255) / 256;
        int np = E_P2G * HEADS, bp = (np + 255) / 256;
        edge_max_kernel<<<bg, 256, 0, stream>>>(g2g_src, g2g_dst, as_g2g, ad_g2g, mk_g2g, E_G2G);
        edge_max_kernel<<<bp, 256, 0, stream>>>(p2g_src, p2g_dst, as_p2g, ad_p2g, mk_p2g, E_P2G);
        edge_den_kernel<<<bg, 256, 0, stream>>>(g2g_src, g2g_dst, as_g2g, ad_g2g, mk_g2g, den_g2g, E_G2G);
        edge_den_kernel<<<bp, 256, 0, stream>>>(p2g_src, p2g_dst, as_p2g, ad_p2g, mk_p2g, den_p2g, E_P2G);
        edge_scatter_kernel<<<E_G2G, 128, 0, stream>>>(g2g_src, g2g_dst, as_g2g, ad_g2g,
                                                       mk_g2g, den_g2g, h_gene, out_g2g);
        edge_scatter_kernel<<<E_P2G, 128, 0, stream>>>(p2g_src, p2g_dst, as_p2g, ad_p2g,
                                                       mk_p2g, den_p2g, h_path, out_p2g);
    }

    // ---- semantic attention (WMMA f32) + fuse + classify ----
    sem_wmma_kernel<<<N_GENE / 16, 256, 0, stream>>>(out_g2g, Wk, bk, wsum);
    sem_wmma_kernel<<<N_GENE / 16, 256, 0, stream>>>(out_p2g, Wk, bk, wsum + HID);
    beta_kernel<<<1, 1, 0, stream>>>(wsum, q, beta);
    final_kernel<<<(N_GENE + 255) / 256, 256, 0, stream>>>(out_g2g, out_p2g, beta,
                                                           W_out, b_out, out);
}